// GCNPredictor_64793876627497
// MI455X (gfx1250) — compile-verified
//
#include <hip/hip_runtime.h>
#include <math.h>

typedef float v2f __attribute__((ext_vector_type(2)));
typedef float v8f __attribute__((ext_vector_type(8)));

#define HDIM 128

// ---------------------------------------------------------------------------
// Degree count: deg[dst[e]] += 1
// ---------------------------------------------------------------------------
__global__ void deg_kernel(const int* __restrict__ dst, float* __restrict__ deg, int nE) {
    int t = blockIdx.x * 256 + threadIdx.x;
    if (t < nE) atomicAdd(&deg[dst[t]], 1.0f);
}

// dinv[n] = rsqrt(deg[n] + 1)   (+1 for self-loop)
__global__ void dinv_kernel(const float* __restrict__ deg, float* __restrict__ dinv, int n) {
    int t = blockIdx.x * 256 + threadIdx.x;
    if (t < n) dinv[t] = rsqrtf(deg[t] + 1.0f);
}

// ---------------------------------------------------------------------------
// GEMM: Out[nrows x 128] = X[nrows x 128] @ W[128 x 128] using fp32 WMMA.
// Block = 256 threads = 8 waves. Block covers 16 rows; wave w covers cols
// [16w, 16w+16). X tile (16x128, 8KB) staged in LDS, shared by all 8 waves.
// 32 x V_WMMA_F32_16X16X4_F32 per tile (K = 128, 4 per instruction).
// Bounds handling is block-uniform (scalar branch) so the hot path has no
// per-lane exec-mask manipulation.
// ---------------------------------------------------------------------------
__global__ __launch_bounds__(256) void gemm_wmma_f32(
    const float* __restrict__ X, const float* __restrict__ W,
    float* __restrict__ Out, int nrows) {
    __shared__ float XT[16 * HDIM];

    const int  row0 = blockIdx.x * 16;
    const bool full = (row0 + 16 <= nrows);    // uniform across block

    if (full) {
        for (int i = threadIdx.x; i < 16 * HDIM; i += 256)
            XT[i] = X[(size_t)row0 * HDIM + i];
    } else {
        for (int i = threadIdx.x; i < 16 * HDIM; i += 256) {
            int r = i >> 7;
            XT[i] = (row0 + r < nrows) ? X[(size_t)row0 * HDIM + i] : 0.0f;
        }
    }
    __syncthreads();

    const int lane = threadIdx.x & 31;
    const int wave = threadIdx.x >> 5;
    const int m    = lane & 15;
    const int koff = (lane >> 4) * 2;      // lanes 0-15 -> K+0/1, lanes 16-31 -> K+2/3
    const int col  = wave * 16 + m;

    v8f acc = {0.f, 0.f, 0.f, 0.f, 0.f, 0.f, 0.f, 0.f};

#pragma unroll 8
    for (int k = 0; k < HDIM; k += 4) {
        v2f a, b;
        // A 16x4 fp32 layout (ISA 7.12.2): lane half picks K pair, VGPR picks element
        a[0] = XT[m * HDIM + k + koff];
        a[1] = XT[m * HDIM + k + koff + 1];
        // B 4x16 fp32: row-striped across lanes; VGPR v + lane-half h -> K = v + 2h
        b[0] = W[(size_t)(k + koff) * HDIM + col];
        b[1] = W[(size_t)(k + koff + 1) * HDIM + col];
        acc = __builtin_amdgcn_wmma_f32_16x16x4_f32(false, a, false, b,
                                                    (short)0, acc, false, false);
    }

    const int mbase = (lane >> 4) * 8;     // D: VGPR r, lane half h -> M = r + 8h
    float* outp = Out + (size_t)(row0 + mbase) * HDIM + col;
    if (full) {
#pragma unroll
        for (int r = 0; r < 8; ++r)
            outp[(size_t)r * HDIM] = acc[r];
    } else {
#pragma unroll
        for (int r = 0; r < 8; ++r)
            if (row0 + mbase + r < nrows) outp[(size_t)r * HDIM] = acc[r];
    }
}

// ---------------------------------------------------------------------------
// Edge scatter: AGG[dst] += dinv[src]*dinv[dst] * H[src]
// One wave per edge; lane handles 4 consecutive features (float4 gather,
// 4 scalar global_atomic_add_f32). 512B coalesced gather per wave; AGG and
// H are L2-resident (51MB each vs 192MB L2).
// ---------------------------------------------------------------------------
__global__ __launch_bounds__(256) void scatter_kernel(
    const float* __restrict__ Hf, const int* __restrict__ src,
    const int* __restrict__ dst, const float* __restrict__ dinv,
    float* __restrict__ AGG, int nE) {
    int t = blockIdx.x * 256 + threadIdx.x;
    int e = t >> 5;
    if (e >= nE) return;
    int lane = t & 31;
    int s = src[e];
    int d = dst[e];
    float nrm = dinv[s] * dinv[d];
    const float4 h = ((const float4*)(Hf + (size_t)s * HDIM))[lane];
    float* base = AGG + (size_t)d * HDIM + lane * 4;
    atomicAdd(base + 0, h.x * nrm);
    atomicAdd(base + 1, h.y * nrm);
    atomicAdd(base + 2, h.z * nrm);
    atomicAdd(base + 3, h.w * nrm);
}

// ---------------------------------------------------------------------------
// Finalize (in place): AGG = relu(AGG + H*dinv^2 + bias)
// One float4 per thread.
// ---------------------------------------------------------------------------
__global__ __launch_bounds__(256) void finalize_kernel(
    float* __restrict__ AGG, const float* __restrict__ Hf,
    const float* __restrict__ dinv, const float* __restrict__ bias, int n) {
    int t = blockIdx.x * 256 + threadIdx.x;   // n*32 float4 chunks
    if (t >= n * 32) return;
    int node = t >> 5;
    int c    = t & 31;
    float di  = dinv[node];
    float di2 = di * di;
    float4 a = ((float4*)AGG)[t];
    float4 h = ((const float4*)Hf)[t];
    float4 b = ((const float4*)bias)[c];
    float4 o;
    o.x = fmaxf(a.x + h.x * di2 + b.x, 0.0f);
    o.y = fmaxf(a.y + h.y * di2 + b.y, 0.0f);
    o.z = fmaxf(a.z + h.z * di2 + b.z, 0.0f);
    o.w = fmaxf(a.w + h.w * di2 + b.w, 0.0f);
    ((float4*)AGG)[t] = o;
}

// ---------------------------------------------------------------------------
// Readout scan: for edges with dst == 0, accumulate H2[src] and count.
// Only ~E/N edges match, so per-thread 128 atomics is negligible work.
// ---------------------------------------------------------------------------
__global__ void readout_kernel(const float* __restrict__ H2,
                               const int* __restrict__ src,
                               const int* __restrict__ dst,
                               float* __restrict__ nsum, float* __restrict__ cnt,
                               int nE) {
    int t = blockIdx.x * 256 + threadIdx.x;
    if (t >= nE) return;
    if (dst[t] == 0) {
        atomicAdd(cnt, 1.0f);
        const float* hs = H2 + (size_t)src[t] * HDIM;
        for (int k = 0; k < HDIM; ++k) atomicAdd(&nsum[k], hs[k]);
    }
}

// ---------------------------------------------------------------------------
// Final dot: out = concat(H2[0], nsum/max(cnt,1)) . fc_w + fc_b
// ---------------------------------------------------------------------------
__global__ __launch_bounds__(256) void final_dot_kernel(
    const float* __restrict__ H2, const float* __restrict__ nsum,
    const float* __restrict__ cnt, const float* __restrict__ fcw,
    const float* __restrict__ fcb, float* __restrict__ out) {
    __shared__ float red[256];
    int i = threadIdx.x;
    float c = (i < HDIM) ? H2[i] : nsum[i - HDIM] / fmaxf(cnt[0], 1.0f);
    red[i] = c * fcw[i];
    __syncthreads();
    for (int s = 128; s > 0; s >>= 1) {
        if (i < s) red[i] += red[i + s];
        __syncthreads();
    }
    if (i == 0) out[0] = red[0] + fcb[0];
}

// ---------------------------------------------------------------------------
extern "C" void kernel_launch(void* const* d_in, const int* in_sizes, int n_in,
                              void* d_out, int out_size, void* d_ws, size_t ws_size,
                              hipStream_t stream) {
    const float* x    = (const float*)d_in[0];
    const int*   ei   = (const int*)d_in[1];   // [2, E] int32
    // d_in[2] = concentration (unused by the reference computation)
    const float* w1   = (const float*)d_in[3];
    const float* b1   = (const float*)d_in[4];
    const float* w2   = (const float*)d_in[5];
    const float* b2   = (const float*)d_in[6];
    const float* fcw  = (const float*)d_in[7];
    const float* fcb  = (const float*)d_in[8];

    const int N = in_sizes[0] / HDIM;
    const int E = in_sizes[1] / 2;
    const int* src = ei;
    const int* dst = ei + E;

    const size_t NH = (size_t)N * HDIM;

    // workspace layout (floats)
    float* A    = (float*)d_ws;     // H1 / H2        (N*H)
    float* B    = A + NH;           // AGG / OUT      (N*H)
    float* deg  = B + NH;           // N
    float* dinv = deg + N;          // N
    float* nsum = dinv + N;         // 128
    float* cnt  = nsum + HDIM;      // 1

    const int gE   = (E + 255) / 256;            // 1 thread / edge
    const int gE32 = ((E * 32) + 255) / 256;     // 1 wave  / edge  (int ok: 51.2M)
    const int gN   = (N + 255) / 256;
    const int gN4  = ((N * 32) + 255) / 256;     // 1 float4 / thread
    const int gT   = (N + 15) / 16;              // GEMM row tiles

    // zero accumulators (graph-capture-safe async memsets)
    hipMemsetAsync(deg, 0, (size_t)N * sizeof(float), stream);
    hipMemsetAsync(B,   0, NH * sizeof(float), stream);
    hipMemsetAsync(nsum, 0, (HDIM + 1) * sizeof(float), stream);

    // degrees + normalization
    deg_kernel<<<gE, 256, 0, stream>>>(dst, deg, E);
    dinv_kernel<<<gN, 256, 0, stream>>>(deg, dinv, N);

    // ---- layer 1 ----
    gemm_wmma_f32<<<gT, 256, 0, stream>>>(x, w1, A, N);             // A = x @ w1
    scatter_kernel<<<gE32, 256, 0, stream>>>(A, src, dst, dinv, B, E);
    finalize_kernel<<<gN4, 256, 0, stream>>>(B, A, dinv, b1, N);    // B = relu(...)

    // ---- layer 2 ----
    gemm_wmma_f32<<<gT, 256, 0, stream>>>(B, w2, A, N);             // A = h1 @ w2
    hipMemsetAsync(B, 0, NH * sizeof(float), stream);
    scatter_kernel<<<gE32, 256, 0, stream>>>(A, src, dst, dinv, B, E);
    finalize_kernel<<<gN4, 256, 0, stream>>>(B, A, dinv, b2, N);    // B = h2

    // ---- readout at node 0 ----
    readout_kernel<<<gE, 256, 0, stream>>>(B, src, dst, nsum, cnt, E);
    final_dot_kernel<<<1, 256, 0, stream>>>(B, nsum, cnt, fcw, fcb, (float*)d_out);
}